// ExpertRouter_86002425135539
// MI455X (gfx1250) — compile-verified
//
#include <hip/hip_runtime.h>
#include <math.h>

// Problem constants (from reference)
#define BB   4
#define SS   2048
#define HH   1024
#define EE   8
#define KK   2
#define CAP  1280                       // ceil(4*2048*1.25/8)
#define NTOK (BB * SS)                  // 8192
#define DISP_ELEMS ((size_t)NTOK * EE * CAP)   // 83,886,080 floats
#define PROBS_OFF  DISP_ELEMS
#define Z_OFF      (PROBS_OFF + (size_t)NTOK * EE)
#define LB_OFF     (Z_OFF + 1)
#define NBLK1      (NTOK / 64)          // 128 blocks for router kernel

typedef __attribute__((ext_vector_type(2))) float v2f;
typedef __attribute__((ext_vector_type(4))) float v4f;
typedef __attribute__((ext_vector_type(8))) float v8f;

// ---------------------------------------------------------------------------
// Kernel 1: router logits via V_WMMA_F32_16X16X4_F32, softmax, top-2,
// per-block deterministic z-loss partial, integer expert counts.
// 128 threads = 4 waves; each wave computes a 16-token x 16-expert tile.
// B (router kernel, 1024x8) is staged once into LDS, zero-padded to N=16 and
// pair-packed so the inner loop is: global_load_b64 (A) + ds_load_b64 (B) +
// v_wmma — branchless, no exec-mask divergence.
// ---------------------------------------------------------------------------
__global__ __launch_bounds__(128)
void moe_router_wmma(const float* __restrict__ hidden,
                     const float* __restrict__ rk,
                     float* __restrict__ out,
                     int*   __restrict__ top2_idx,
                     float* __restrict__ top2_prob,
                     float* __restrict__ zpart,
                     unsigned int* __restrict__ counts) {
    // bpair[k/2][n][k&1] = B element (K=k, N=n), zero for n >= 8.  64 KB.
    __shared__ float bpair[HH / 2][16][2];
    __shared__ float tile[4][16][17];   // +1 pad vs LDS bank conflicts
    __shared__ float zred[64];

    const int tid  = threadIdx.x;
    const int wave = tid >> 5;
    const int lane = tid & 31;
    const int half = lane >> 4;         // 0: K=0..1 side, 1: K=2..3 side
    const int lm   = lane & 15;
    const int tok0 = (blockIdx.x * 4 + wave) * 16;

    // Stage zero-padded B into LDS (one-time, 16384 floats / 128 threads).
    for (int idx = tid; idx < HH * 16; idx += 128) {
        const int k = idx >> 4;
        const int n = idx & 15;
        bpair[k >> 1][n][k & 1] = (n < EE) ? rk[(size_t)k * EE + n] : 0.0f;
    }
    __syncthreads();

    // A-matrix row for this lane (M = lm), starting at its K half.
    const float* arow = hidden + (size_t)(tok0 + lm) * HH + half * 2;

    v8f c = {};
    #pragma unroll 8
    for (int k0 = 0; k0 < HH; k0 += 4) {
        const v2f a = *(const v2f*)(arow + k0);                     // K=k0+2h, +1
        const v2f b = *(const v2f*)(&bpair[(k0 >> 1) + half][lm][0]);
        // D = A(16x4) * B(4x16) + C, full f32
        c = __builtin_amdgcn_wmma_f32_16x16x4_f32(false, a, false, b,
                                                  (short)0, c, false, false);
    }

    // C layout: VGPR r holds M = r + 8*half, N = lm  -> spill tile to LDS
    #pragma unroll
    for (int r = 0; r < 8; ++r)
        tile[wave][r + half * 8][lm] = c[r];
    __syncthreads();

    if (tid < 64) {
        const int w   = tid >> 4;
        const int row = tid & 15;
        const int t   = (blockIdx.x * 4 + w) * 16 + row;

        float l[EE];
        float m = -INFINITY;
        #pragma unroll
        for (int e = 0; e < EE; ++e) { l[e] = tile[w][row][e]; m = fmaxf(m, l[e]); }

        float p[EE];
        float sum = 0.0f;
        #pragma unroll
        for (int e = 0; e < EE; ++e) { p[e] = expf(l[e] - m); sum += p[e]; }
        const float inv = 1.0f / sum;

        int   i1 = 0, i2 = 0;
        float p1 = -1.0f, p2 = -1.0f;
        #pragma unroll
        for (int e = 0; e < EE; ++e) {
            const float pe = p[e] * inv;
            out[PROBS_OFF + (size_t)t * EE + e] = pe;
            if (pe > p1)      { p2 = p1; i2 = i1; p1 = pe; i1 = e; }
            else if (pe > p2) { p2 = pe; i2 = e; }
        }
        top2_idx[2 * t + 0]  = i1;
        top2_idx[2 * t + 1]  = i2;
        top2_prob[2 * t + 0] = p1;
        top2_prob[2 * t + 1] = p2;
        atomicAdd(&counts[i1], 1u);     // integer adds: order-insensitive
        atomicAdd(&counts[i2], 1u);

        const float lz = m + logf(sum); // logsumexp
        zred[tid] = lz * lz;
    }
    __syncthreads();

    if (tid == 0) {                     // fixed-order partial: deterministic
        float s = 0.0f;
        for (int i = 0; i < 64; ++i) s += zred[i];
        zpart[blockIdx.x] = s;
    }
}

// ---------------------------------------------------------------------------
// Kernel 2: per-(b,e) exclusive prefix count along S via wave32 ballot.
// 32 blocks (one per b,e chain), 32 threads (one wave).
// ---------------------------------------------------------------------------
__global__ __launch_bounds__(32)
void moe_scan(const int* __restrict__ top2_idx, int* __restrict__ pos_k) {
    const int b = blockIdx.x >> 3;
    const int e = blockIdx.x & 7;
    const int lane = threadIdx.x;

    unsigned carry = 0;
    for (int chunk = 0; chunk < SS / 32; ++chunk) {
        const int s = chunk * 32 + lane;
        const int t = b * SS + s;
        const int i0 = top2_idx[2 * t + 0];
        const int i1 = top2_idx[2 * t + 1];
        const bool bit = (i0 == e) || (i1 == e);
        const unsigned m32 = (unsigned)__ballot(bit);  // wave32: low 32 bits
        const unsigned pre = __popc(m32 & ((1u << lane) - 1u));
        if (bit) {
            const int k = (i0 == e) ? 0 : 1;
            pos_k[2 * t + k] = (int)(carry + pre);
        }
        carry += __popc(m32);
    }
}

// ---------------------------------------------------------------------------
// Kernel 3: streaming zero-fill of the 335.5 MB dispatch tensor.
// 128-bit nontemporal stores; this is the bandwidth-dominant op (~14 us).
// ---------------------------------------------------------------------------
__global__ __launch_bounds__(256)
void moe_zero_dispatch(float* __restrict__ out) {
    const size_t n4 = DISP_ELEMS / 4;
    const size_t stride = (size_t)gridDim.x * blockDim.x;
    v4f* __restrict__ o4 = (v4f*)out;
    const v4f z = {0.0f, 0.0f, 0.0f, 0.0f};
    for (size_t i = (size_t)blockIdx.x * blockDim.x + threadIdx.x; i < n4; i += stride)
        __builtin_nontemporal_store(z, &o4[i]);
}

// ---------------------------------------------------------------------------
// Kernel 4: scatter the 16384 selected probs into dispatch.
// (dropped slots write nothing: reference writes 0.0 into a zeroed buffer)
// ---------------------------------------------------------------------------
__global__ __launch_bounds__(256)
void moe_scatter(const int* __restrict__ top2_idx,
                 const float* __restrict__ top2_prob,
                 const int* __restrict__ pos_k,
                 float* __restrict__ out) {
    const int i = blockIdx.x * 256 + threadIdx.x;
    if (i >= NTOK * KK) return;
    const int pos = pos_k[i];
    if (pos < CAP) {
        const int t = i >> 1;
        const int e = top2_idx[i];
        out[((size_t)t * EE + e) * CAP + pos] = top2_prob[i];
    }
}

// ---------------------------------------------------------------------------
// Kernel 5: finalize scalars (fixed-order sums: deterministic).
// ---------------------------------------------------------------------------
__global__ void moe_finalize(const float* __restrict__ zpart,
                             const unsigned int* __restrict__ counts,
                             float* __restrict__ out) {
    if (threadIdx.x == 0 && blockIdx.x == 0) {
        float s = 0.0f;
        for (int i = 0; i < NBLK1; ++i) s += zpart[i];
        out[Z_OFF] = s / (float)NTOK;

        float lb = 0.0f;
        for (int e = 0; e < EE; ++e) {
            const float f = (float)counts[e] / (float)NTOK;
            lb += f * logf(f * (float)EE);
        }
        out[LB_OFF] = lb;
    }
}

__global__ void moe_init_ws(unsigned int* __restrict__ counts) {
    if (threadIdx.x < EE) counts[threadIdx.x] = 0u;
}

// ---------------------------------------------------------------------------
extern "C" void kernel_launch(void* const* d_in, const int* in_sizes, int n_in,
                              void* d_out, int out_size, void* d_ws, size_t ws_size,
                              hipStream_t stream) {
    const float* hidden = (const float*)d_in[0];   // [4,2048,1024] f32
    const float* rk     = (const float*)d_in[1];   // [1024,8] f32
    float* out = (float*)d_out;

    // Workspace layout
    char* ws = (char*)d_ws;
    int*          top2_idx  = (int*)(ws + 0);                 // 16384 ints
    float*        top2_prob = (float*)(ws + 64 * 1024);       // 16384 floats
    int*          pos_k     = (int*)(ws + 128 * 1024);        // 16384 ints
    float*        zpart     = (float*)(ws + 192 * 1024);      // 128 floats
    unsigned int* counts    = (unsigned int*)(ws + 193 * 1024); // 8 uints

    moe_init_ws<<<1, 32, 0, stream>>>(counts);

    // Bandwidth-dominant zero-fill first (stream is serial; this is the
    // largest single op either way).
    moe_zero_dispatch<<<2048, 256, 0, stream>>>(out);

    moe_router_wmma<<<NBLK1, 128, 0, stream>>>(hidden, rk, out, top2_idx,
                                               top2_prob, zpart, counts);

    moe_scan<<<BB * EE, 32, 0, stream>>>(top2_idx, pos_k);

    moe_scatter<<<(NTOK * KK + 255) / 256, 256, 0, stream>>>(top2_idx, top2_prob,
                                                             pos_k, out);

    moe_finalize<<<1, 32, 0, stream>>>(zpart, counts, out);
}